// Model_51453708206373
// MI455X (gfx1250) — compile-verified
//
#include <hip/hip_runtime.h>
#include <stdint.h>

// Greedy NMS for MI455X (gfx1250, wave32).
// d_in: [0]=boxes (4N f32), [1]=scores (N f32), [2]=max_output_size (i32),
//       [3]=iou_threshold (f32), [4]=scores_threshold (f32), [5]=pad flag
// d_out: int32[maxo] selected indices (zero padded) + int32 count, maxo = out_size-1.
//
// NMS has no inner-product structure (IoU = min/max/mul, selection = argmax),
// so WMMA does not apply. The CDNA5-specific paths used instead:
//   - global_load_async_to_lds_b128 + s_wait_asynccnt (double-buffered pipeline)
//   - global_prefetch_b8 via __builtin_prefetch
//   - wave32 ds-swizzle reductions + non-returning global_atomic_max_u64

#define NEG_INF (-__builtin_inff())
#define BLOCK 256
#define NWAVE (BLOCK / 32)

struct NmsState {
  unsigned long long nextKey;   // packed (score_bits<<32) | (0xFFFFFFFF - idx)
  float wx1, wy1, wx2, wy2;     // winner box
  float wArea;
  int   winIdx;
  int   winValid;
  int   selCount;
};

static __device__ __forceinline__ unsigned long long packKey(float s, unsigned i) {
  return (((unsigned long long)__float_as_uint(s)) << 32) |
         (unsigned long long)(0xFFFFFFFFu - i);
}

static __device__ __forceinline__ unsigned long long shflxor64(unsigned long long v, int mask) {
  unsigned lo = (unsigned)(v & 0xFFFFFFFFull);
  unsigned hi = (unsigned)(v >> 32);
  lo = __shfl_xor(lo, mask, 32);   // wave32: width 32
  hi = __shfl_xor(hi, mask, 32);
  return (((unsigned long long)hi) << 32) | (unsigned long long)lo;
}

// wave -> block (LDS) -> one non-returning u64 atomic per block
static __device__ __forceinline__ void blockReduceAtomicMax(unsigned long long key,
                                                            unsigned long long* dst) {
  __shared__ unsigned long long wbest[NWAVE];
#pragma unroll
  for (int m = 16; m >= 1; m >>= 1) {
    unsigned long long o = shflxor64(key, m);
    if (o > key) key = o;
  }
  const unsigned lane = threadIdx.x & 31u;
  const unsigned wid  = threadIdx.x >> 5;
  if (lane == 0) wbest[wid] = key;
  __syncthreads();
  if (threadIdx.x == 0) {
    unsigned long long k = wbest[0];
#pragma unroll
    for (int w = 1; w < NWAVE; ++w)
      if (wbest[w] > k) k = wbest[w];
    if (k != 0ull) atomicMax(dst, k);   // return value unused -> non-returning atomic
  }
}

__global__ void nms_reset(NmsState* st) {
  if (threadIdx.x == 0) {
    st->nextKey = 0ull;
    st->wx1 = st->wy1 = st->wx2 = st->wy2 = 0.f;
    st->wArea = 0.f;
    st->winIdx = -1;
    st->winValid = 0;
    st->selCount = 0;
  }
}

__global__ void __launch_bounds__(BLOCK)
nms_init(const float* __restrict__ scores,
         const float* __restrict__ scoThrP,
         float* __restrict__ live,
         NmsState* __restrict__ st, int n) {
  const float thr = *scoThrP;
  unsigned long long best = 0ull;
  const int stride = gridDim.x * blockDim.x;
  for (int i = blockIdx.x * blockDim.x + threadIdx.x; i < n; i += stride) {
    __builtin_prefetch(&scores[i + stride], 0, 0);   // -> global_prefetch_b8 (speculative)
    float s = scores[i];
    bool keep = s > thr;
    live[i] = keep ? s : NEG_INF;
    if (keep) {
      unsigned long long k = packKey(s, (unsigned)i);
      if (k > best) best = k;
    }
  }
  blockReduceAtomicMax(best, &st->nextKey);
}

__global__ void nms_finalize(const float4* __restrict__ boxes,
                             int* __restrict__ sel,
                             NmsState* st) {
  if (threadIdx.x != 0) return;
  unsigned long long k = st->nextKey;
  if (k != 0ull) {  // a live box with score > threshold exists
    unsigned idx = 0xFFFFFFFFu - (unsigned)(k & 0xFFFFFFFFull);
    float4 b = boxes[idx];
    st->wx1 = b.x; st->wy1 = b.y; st->wx2 = b.z; st->wy2 = b.w;
    st->wArea = (b.z - b.x) * (b.w - b.y);
    st->winIdx = (int)idx;
    st->winValid = 1;
    sel[st->selCount] = (int)idx;
    st->selCount = st->selCount + 1;
  } else {
    st->winValid = 0;
  }
  st->nextKey = 0ull;
}

// Fused kill-vs-winner + argmax-for-next-round.
// Boxes are staged through LDS with the CDNA5 async copy path (ASYNCcnt),
// double-buffered: issue chunk k+1, wait ASYNCcnt<=1, consume chunk k.
__global__ void __launch_bounds__(BLOCK)
nms_suppress(const float4* __restrict__ boxes,
             const float* __restrict__ iouThrP,
             float* __restrict__ live,
             NmsState* __restrict__ st, int n) {
  __shared__ float4 tile[2 * BLOCK];

  const float iouThr   = *iouThrP;
  const int   winValid = st->winValid;
  const int   winIdx   = st->winIdx;
  const float wx1 = st->wx1, wy1 = st->wy1, wx2 = st->wx2, wy2 = st->wy2;
  const float wArea = st->wArea;

  // Generic LDS pointer: low 32 bits are the LDS byte offset on gfx1250.
  const unsigned ldsBase = (unsigned)(uintptr_t)&tile[0];
  const unsigned long long boxBase = (unsigned long long)(uintptr_t)boxes;
  const int tid = threadIdx.x;

  // Issue one async b128 (= one box) per lane for chunk starting at cb.
  // Index clamped to n-1 so all lanes can issue (no OOB, no divergence).
  auto issue = [&](long cb, int buf) {
    long li = cb + (long)tid;
    if (li > (long)n - 1) li = (long)n - 1;
    unsigned dst  = ldsBase + (unsigned)(buf * BLOCK + tid) * 16u;
    unsigned voff = (unsigned)li * 16u;
    asm volatile("global_load_async_to_lds_b128 %0, %1, %2 offset:0"
                 :: "v"(dst), "v"(voff), "s"(boxBase)
                 : "memory");
  };

  unsigned long long best = 0ull;
  const long cstride = (long)gridDim.x * (long)BLOCK;
  long cb0 = (long)blockIdx.x * (long)BLOCK;

  if (winValid) {
    if (cb0 < n) issue(cb0, 0);                       // prime the pipeline
    int buf = 0;
    for (long cb = cb0; cb < n; cb += cstride, buf ^= 1) {
      const long nb = cb + cstride;
      const int  i  = (int)(cb + tid);
      float s = (i < n) ? live[i] : NEG_INF;

      if (nb < n) {
        issue(nb, buf ^ 1);                           // overlap next chunk's copy
        asm volatile("s_wait_asynccnt 0x1" ::: "memory");
      } else {
        asm volatile("s_wait_asynccnt 0x0" ::: "memory");
      }

      if (s != NEG_INF) {
        float4 b = tile[buf * BLOCK + tid];
        float ix1 = fmaxf(wx1, b.x), iy1 = fmaxf(wy1, b.y);
        float ix2 = fminf(wx2, b.z), iy2 = fminf(wy2, b.w);
        float inter = fmaxf(ix2 - ix1, 0.f) * fmaxf(iy2 - iy1, 0.f);
        float area  = (b.z - b.x) * (b.w - b.y);
        float uni   = fmaxf(wArea + area - inter, 1e-6f);
        float iou   = inter / uni;
        if (iou >= iouThr || i == winIdx) {
          live[i] = NEG_INF;
          s = NEG_INF;
        }
      }
      if (s != NEG_INF) {
        unsigned long long k = packKey(s, (unsigned)i);
        if (k > best) best = k;
      }
    }
  } else {
    // No winner this round: nothing can change; argmax over live only.
    for (long cb = cb0; cb < n; cb += cstride) {
      const int i = (int)(cb + tid);
      if (i < n) {
        float s = live[i];
        if (s != NEG_INF) {
          unsigned long long k = packKey(s, (unsigned)i);
          if (k > best) best = k;
        }
      }
    }
  }
  blockReduceAtomicMax(best, &st->nextKey);
}

__global__ void nms_writeout(const int* __restrict__ sel,
                             const NmsState* __restrict__ st,
                             int* __restrict__ out, int maxo) {
  int j = blockIdx.x * blockDim.x + threadIdx.x;
  int cnt = st->selCount;
  if (j < maxo) out[j] = (j < cnt) ? sel[j] : 0;
  if (j == maxo) out[maxo] = cnt;
}

extern "C" void kernel_launch(void* const* d_in, const int* in_sizes, int n_in,
                              void* d_out, int out_size, void* d_ws, size_t ws_size,
                              hipStream_t stream) {
  (void)n_in; (void)ws_size;
  const float4* boxes  = (const float4*)d_in[0];
  const float*  scores = (const float*)d_in[1];
  const float*  iouThr = (const float*)d_in[3];
  const float*  scoThr = (const float*)d_in[4];

  const int n = in_sizes[1];             // number of boxes
  int maxo = out_size - 1;               // selected indices + trailing count
  if (maxo < 0) maxo = 0;

  char* ws = (char*)d_ws;
  NmsState* st = (NmsState*)ws;                         // state at offset 0
  int* sel = (int*)(ws + 256);                          // selected indices
  size_t selBytes = (((size_t)maxo * sizeof(int)) + 255) & ~(size_t)255;
  float* live = (float*)(ws + 256 + selBytes);          // N live scores

  int blocks = 1024;                                    // 256K threads, grid-stride
  int needed = (n + BLOCK - 1) / BLOCK;
  if (needed < 1) needed = 1;
  if (blocks > needed) blocks = needed;

  nms_reset<<<1, 32, 0, stream>>>(st);
  nms_init<<<blocks, BLOCK, 0, stream>>>(scores, scoThr, live, st, n);
  for (int t = 0; t < maxo; ++t) {
    nms_finalize<<<1, 32, 0, stream>>>(boxes, sel, st);
    if (t + 1 < maxo)
      nms_suppress<<<blocks, BLOCK, 0, stream>>>(boxes, iouThr, live, st, n);
  }
  nms_writeout<<<(out_size + BLOCK - 1) / BLOCK, BLOCK, 0, stream>>>(sel, st, (int*)d_out, maxo);
}